// GATLayer_31533649887298
// MI455X (gfx1250) — compile-verified
//
#include <hip/hip_runtime.h>
#include <hip/hip_bf16.h>

typedef __attribute__((ext_vector_type(2))) float v2f;
typedef __attribute__((ext_vector_type(8))) float v8f;

#define GAT_NEG_INF (-9000000000000000.0f)

constexpr int kB   = 8;
constexpr int kN   = 1024;
constexpr int kDin = 128;
constexpr int kH   = 8;
constexpr int kD   = 16;
constexpr int kHD  = 128;   // H*D
constexpr float kAlpha = 0.2f;

__device__ __forceinline__ float leaky(float x) {
    // alpha < 1  =>  leaky_relu(x) == max(x, alpha*x)
    return fmaxf(x, kAlpha * x);
}

// Raw cross-lane ops (avoid HIP __shfl's clamp/cndmask overhead).
template <int IMM>
__device__ __forceinline__ float swzf(float v) {   // ds_swizzle, xor pattern
    return __int_as_float(__builtin_amdgcn_ds_swizzle(__float_as_int(v), IMM));
}
__device__ __forceinline__ float bpermf(int byteAddr, float v) {
    return __int_as_float(__builtin_amdgcn_ds_bpermute(byteAddr, __float_as_int(v)));
}
// xor-mask immediates: offset = (xor<<10) | (or<<5) | and(0x1f)
#define SWZ_X1  0x041f
#define SWZ_X2  0x081f
#define SWZ_X4  0x101f
#define SWZ_X8  0x201f
#define SWZ_X16 0x401f

// ---------------------------------------------------------------------------
// Phase 1: Wh = h @ W (8192x128 @ 128x128) via V_WMMA_F32_16X16X4_F32,
// plus attention logits s_i = Wh·a_src, s_j = Wh·a_dst.
// Wh is written in PAIRED layout: WhP[(row/2)*128 + col][2] with element
// [row&1] -> phase 2 fetches the (j, j+1) K-pair with one b64 load.
// ---------------------------------------------------------------------------
__global__ __launch_bounds__(256) void gat_phase1(
    const float* __restrict__ h, const float* __restrict__ W,
    const float* __restrict__ a, float* __restrict__ WhP,
    float* __restrict__ s_i, float* __restrict__ s_j)
{
    __shared__ float hT[16 * 132];          // stride 132 -> conflict-free A reads

    const int g    = blockIdx.x;            // 16-row tile index (0..511)
    const int tid  = threadIdx.x;
    const int head = tid >> 5;               // wave id == head == n-tile
    const int lane = tid & 31;
    const int r16  = lane & 15;
    const int half = lane >> 4;

    {   // stage h tile (16 x 128), coalesced float4
        const float4* src = (const float4*)(h + (size_t)g * 16 * kDin);
        #pragma unroll
        for (int it = 0; it < 2; ++it) {
            int idx  = it * 256 + tid;
            int row  = idx >> 5;
            int col4 = idx & 31;
            float4 v = src[idx];
            *(float4*)&hT[row * 132 + col4 * 4] = v;
        }
    }
    __syncthreads();

    const int n0 = head * 16;
    v8f c = {};
    #pragma unroll 4
    for (int k = 0; k < kDin; k += 4) {
        const int ka = k + 2 * half;
        v2f av, bv;
        av[0] = hT[r16 * 132 + ka];
        av[1] = hT[r16 * 132 + ka + 1];
        bv[0] = W[(size_t)ka * kHD + n0 + r16];
        bv[1] = W[(size_t)(ka + 1) * kHD + n0 + r16];
        c = __builtin_amdgcn_wmma_f32_16x16x4_f32(false, av, false, bv,
                                                  (short)0, c, false, false);
    }

    // Store Wh tile in paired layout.
    #pragma unroll
    for (int r = 0; r < 8; ++r) {
        int row = g * 16 + r + half * 8;
        WhP[((size_t)(row >> 1) * kHD + n0 + r16) * 2 + (row & 1)] = c[r];
    }

    // s_i / s_j: reduce over D=16 within each half-wave (xor 1,2,4,8).
    const float asrc = a[head * 2 * kD + r16];
    const float adst = a[head * 2 * kD + kD + r16];
    #pragma unroll
    for (int r = 0; r < 8; ++r) {
        float vs = c[r] * asrc;
        float vd = c[r] * adst;
        vs += swzf<SWZ_X1>(vs);  vd += swzf<SWZ_X1>(vd);
        vs += swzf<SWZ_X2>(vs);  vd += swzf<SWZ_X2>(vd);
        vs += swzf<SWZ_X4>(vs);  vd += swzf<SWZ_X4>(vd);
        vs += swzf<SWZ_X8>(vs);  vd += swzf<SWZ_X8>(vd);
        if (r16 == 0) {
            int row = g * 16 + r + half * 8;
            int bb = row >> 10, nn = row & (kN - 1);
            s_i[((size_t)bb * kH + head) * kN + nn] = vs;
            s_j[((size_t)bb * kH + head) * kN + nn] = vd;
        }
    }
}

// ---------------------------------------------------------------------------
// Phase 2: fused masked softmax + (attn @ Wh).
// Block = (b, 16-row tile); 8 waves = 8 heads share one LDS adj-mask tile.
// Additive mask stored paired: adjM[(j>>1)*32 + 2*((r + (j>>2))&15) + (j&1)],
// value 0.0f (edge) or NEG_INF (no edge). Rotation swizzle makes the hot
// b64 read (r varies per lane, j fixed per half) bank-conflict free.
// Row max uses the unmasked bound m = leaky(s_i + max_j s_j) (softmax is
// shift-invariant; leaky is monotone), eliminating a full masked-max sweep.
// ---------------------------------------------------------------------------
__global__ __launch_bounds__(256) void gat_phase2(
    const int* __restrict__ adj, const float* __restrict__ WhP,
    const float* __restrict__ s_i, const float* __restrict__ s_j,
    float* __restrict__ out)
{
    __shared__ float adjM[kN * 16];          // 64 KB

    const int blk  = blockIdx.x;             // 0..511
    const int b    = blk >> 6;
    const int i0   = (blk & 63) * 16;
    const int tid  = threadIdx.x;
    const int head = tid >> 5;
    const int lane = tid & 31;
    const int r16  = lane & 15;
    const int half = lane >> 4;

    {   // stage additive mask, coalesced int4 global loads, paired f2 stores
        const int4* src = (const int4*)(adj + ((size_t)b * kN + i0) * kN);
        #pragma unroll
        for (int it = 0; it < 16; ++it) {
            int idx = it * 256 + tid;        // int4 index: 256 per row
            int row = idx >> 8;
            int col = (idx & 255) * 4;       // multiple of 4
            int4 v  = src[idx];
            int rot0 = (col >> 2) & 15;      // same for col..col+3
            int base = (col >> 1) * 32 + 2 * ((row + rot0) & 15);
            float2 lo, hi;
            lo.x = v.x ? 0.0f : GAT_NEG_INF;
            lo.y = v.y ? 0.0f : GAT_NEG_INF;
            hi.x = v.z ? 0.0f : GAT_NEG_INF;
            hi.y = v.w ? 0.0f : GAT_NEG_INF;
            *(float2*)&adjM[base]      = lo;     // j = col, col+1
            *(float2*)&adjM[base + 32] = hi;     // j = col+2, col+3
        }
    }
    __syncthreads();

    const float* sI = s_i + ((size_t)b * kH + head) * kN;
    const float* sJ = s_j + ((size_t)b * kH + head) * kN;
    const float si_l = sI[i0 + r16];         // this lane's A-row logit

    // Global max of s_j for this head (upper bound for every row's max).
    float mv = GAT_NEG_INF;
    {
        const float4* sJ4 = (const float4*)sJ;
        #pragma unroll
        for (int k = 0; k < 8; ++k) {
            float4 v = sJ4[k * 32 + lane];
            mv = fmaxf(mv, fmaxf(fmaxf(v.x, v.y), fmaxf(v.z, v.w)));
        }
        mv = fmaxf(mv, swzf<SWZ_X16>(mv));
        mv = fmaxf(mv, swzf<SWZ_X8>(mv));
        mv = fmaxf(mv, swzf<SWZ_X4>(mv));
        mv = fmaxf(mv, swzf<SWZ_X2>(mv));
        mv = fmaxf(mv, swzf<SWZ_X1>(mv));
    }
    const float m_l = leaky(si_l + mv);      // per-row stable shift

    // Main sweep: P = exp(score - m) built directly in WMMA A layout,
    // fused row-sum, C += P @ Wh with fp32 WMMA (K=4 per step).
    const float* WhPB = WhP + (size_t)b * kN * kHD + head * 32;  // paired base
    v8f c = {};
    float psum = 0.0f;
    for (int chunk = 0; chunk < 32; ++chunk) {
        const int jb = chunk * 32;
        #pragma unroll 4
        for (int kk = 0; kk < 8; ++kk) {
            const int j0  = jb + kk * 4 + 2 * half;   // even; K-pair (j0, j0+1)
            const int q0  = j0 >> 1;
            const int rot = (j0 >> 2) & 15;
            const int idx = q0 * 32 + 2 * ((r16 + rot) & 15);
            float2 ad = *(const float2*)&adjM[idx];                 // mask pair
            float2 sj = *(const float2*)(sJ + j0);                  // logit pair
            float2 wv = *(const float2*)(WhPB + (size_t)q0 * 256 + r16 * 2);
            float t0 = leaky(si_l + sj.x) + ad.x;
            float t1 = leaky(si_l + sj.y) + ad.y;
            float p0 = __expf(t0 - m_l);
            float p1 = __expf(t1 - m_l);
            psum += p0 + p1;
            v2f av, bv;
            av[0] = p0;   av[1] = p1;
            bv[0] = wv.x; bv[1] = wv.y;
            c = __builtin_amdgcn_wmma_f32_16x16x4_f32(false, av, false, bv,
                                                      (short)0, c, false, false);
        }
    }

    // Row sums: lanes L and L+16 cover disjoint j's of row r16.
    psum += swzf<SWZ_X16>(psum);
    const float inv_l = 1.0f / fmaxf(psum, 1e-30f);
    const int baddr = half * 32;             // bpermute byte base (rows half*8..)
    #pragma unroll
    for (int r = 0; r < 8; ++r) {
        float sc = bpermf(baddr + 4 * r, inv_l);   // 1/l for this C row
        int row = i0 + r + half * 8;
        out[((size_t)b * kN + row) * kHD + head * 16 + r16] = c[r] * sc;
    }
}

// ---------------------------------------------------------------------------
extern "C" void kernel_launch(void* const* d_in, const int* in_sizes, int n_in,
                              void* d_out, int out_size, void* d_ws, size_t ws_size,
                              hipStream_t stream) {
    const float* h   = (const float*)d_in[0];
    const int*   adj = (const int*)d_in[1];
    const float* W   = (const float*)d_in[2];
    const float* a   = (const float*)d_in[3];
    float* out = (float*)d_out;

    float* WhP = (float*)d_ws;                       // 8192*128 floats (4 MB)
    float* s_i = WhP + (size_t)kB * kN * kHD;        // 64K floats
    float* s_j = s_i + (size_t)kB * kH * kN;         // 64K floats

    gat_phase1<<<dim3((kB * kN) / 16), 256, 0, stream>>>(h, W, a, WhP, s_i, s_j);
    gat_phase2<<<dim3(kB * (kN / 16)), 256, 0, stream>>>(adj, WhP, s_i, s_j, out);
}